// LorentzEquivariantAdapter_27659589386271
// MI455X (gfx1250) — compile-verified
//
#include <hip/hip_runtime.h>
#include <hip/hip_bf16.h>

// LorentzEquivariantAdapter fused kernel for gfx1250 (MI455X), fp32 WMMA path.
//
// Per 32-token tile (64 threads = 2 waves, ~146 KB LDS -> 2 WGs per WGP so
// HBM staging of one WG overlaps compute of the other):
//   stage x tile -> LDS via global_load_async_to_lds_b128 (ASYNCcnt path)
//   LN stats (mu, rsig) per token; LN affine folded into down-GEMM epilogue
//   down:  C = x_tile @ WgB   (Wg = gamma .* w_down, pre-swizzled frags) + fold + GELU
//   mid :  h2 = GELU(h1 @ M + b0)  (to_mv/equi/from_mv fused into 48x48 M)
//   up  :  out = x + scale * (h2 @ w_upT)   (residual read from LDS)

typedef __attribute__((ext_vector_type(2))) float v2f;
typedef __attribute__((ext_vector_type(8))) float v8f;
typedef int v4i_vs __attribute__((vector_size(16)));

#define DN 1024
#define RB 48
#define TOK_PER_WG 32
#define XPITCH 1028   // 1028 mod 64 == 4 -> conflict-free 16-row strided b64 loads
#define HPITCH 52     // 52 = 4*13 -> conflict-free as well

#if defined(__gfx1250__) && __has_builtin(__builtin_amdgcn_global_load_async_to_lds_b128)
#define USE_ASYNC_LDS 1
#else
#define USE_ASYNC_LDS 0
#endif

__device__ __forceinline__ float gelu_exact(float z) {
  return 0.5f * z * (1.0f + erff(z * 0.70710678118654752440f));
}

// ---------------- pack kernels (run once per launch; tiny) ----------------

// down B frags: B[k][n] = gamma[k]*w_down[n][k]; frag idx ((nt*256+s)*32+lane)
__global__ void pack_down(const float* __restrict__ gamma,
                          const float* __restrict__ w_down,
                          v2f* __restrict__ wgB) {
  int idx = blockIdx.x * blockDim.x + threadIdx.x;
  if (idx >= 3 * 256 * 32) return;
  int lane = idx & 31;
  int s = (idx >> 5) & 255;
  int nt = idx >> 13;
  int k = 4 * s + 2 * (lane >> 4);
  int n = nt * 16 + (lane & 15);
  v2f v;
  v.x = gamma[k] * w_down[n * DN + k];
  v.y = gamma[k + 1] * w_down[n * DN + k + 1];
  wgB[idx] = v;
}

// up B frags: B[r][d] = w_up[d][r]; frag idx ((nt*12+s)*32+lane), nt in [0,64)
__global__ void pack_up(const float* __restrict__ w_up, v2f* __restrict__ upB) {
  int idx = blockIdx.x * blockDim.x + threadIdx.x;
  if (idx >= 64 * 12 * 32) return;
  int lane = idx & 31;
  int s = (idx >> 5) % 12;
  int nt = idx / 384;
  int r = 4 * s + 2 * (lane >> 4);
  int d = nt * 16 + (lane & 15);
  v2f v;
  v.x = w_up[d * RB + r];
  v.y = w_up[d * RB + r + 1];
  upB[idx] = v;
}

__device__ __forceinline__ float grade_w(const float* we, int j) {
  int g = (j == 0) ? 0 : (j < 5) ? 1 : (j < 11) ? 2 : (j < 15) ? 3 : 4;
  return we[g];
}

// fused middle matrix M[r][r'] = sum_j g[j]*w_to_mv[j][r]*w_from_mv[r'][j]
// frag idx ((nt*12+s)*32+lane), nt in [0,3); plus b0[r'] = b_equi*w_from_mv[r'][0]
__global__ void pack_mid(const float* __restrict__ w_to_mv,
                         const float* __restrict__ w_from_mv,
                         const float* __restrict__ w_equi,
                         const float* __restrict__ b_equi,
                         v2f* __restrict__ midB, float* __restrict__ b0) {
  int idx = blockIdx.x * blockDim.x + threadIdx.x;
  if (idx < 48) b0[idx] = b_equi[0] * w_from_mv[idx * 16 + 0];
  if (idx >= 3 * 12 * 32) return;
  int lane = idx & 31;
  int s = (idx >> 5) % 12;
  int nt = idx / 384;
  int r = 4 * s + 2 * (lane >> 4);
  int n = nt * 16 + (lane & 15);
  float m0 = 0.f, m1 = 0.f;
  for (int j = 0; j < 16; ++j) {
    float g = grade_w(w_equi, j);
    float fm = w_from_mv[n * 16 + j];
    m0 += g * w_to_mv[j * RB + r] * fm;
    m1 += g * w_to_mv[j * RB + r + 1] * fm;
  }
  v2f v; v.x = m0; v.y = m1;
  midB[idx] = v;
}

// s1[r] = sum_d beta[d]*w_down[r][d];  s2[r] = sum_d gamma[d]*w_down[r][d]
__global__ void pack_sums(const float* __restrict__ gamma,
                          const float* __restrict__ beta,
                          const float* __restrict__ w_down,
                          float* __restrict__ s1, float* __restrict__ s2) {
  int r = threadIdx.x;
  if (r >= RB) return;
  float a = 0.f, b = 0.f;
  for (int d = 0; d < DN; ++d) {
    float w = w_down[r * DN + d];
    a += beta[d] * w;
    b += gamma[d] * w;
  }
  s1[r] = a;
  s2[r] = b;
}

// ---------------- main fused kernel ----------------

__global__ __launch_bounds__(64) void adapter_main(
    const float* __restrict__ x, const float* __restrict__ s1g,
    const float* __restrict__ s2g, const v2f* __restrict__ wgB,
    const v2f* __restrict__ midB, const float* __restrict__ b0g,
    const v2f* __restrict__ upB, const float* __restrict__ scale_p,
    float* __restrict__ out) {
  extern __shared__ float lds[];
  float* s_x = lds;                               // 32*1028
  float* s_h1 = s_x + TOK_PER_WG * XPITCH;        // 32*52
  float* s_h2 = s_h1 + TOK_PER_WG * HPITCH;       // 32*52
  float* s_mu = s_h2 + TOK_PER_WG * HPITCH;       // 32
  float* s_rs = s_mu + TOK_PER_WG;                // 32
  float* s_ps = s_rs + TOK_PER_WG;                // 64
  float* s_pq = s_ps + 64;                        // 64

  const int t = threadIdx.x;
  const int lane = t & 31;
  const int wave = t >> 5;
  const long tile = blockIdx.x;
  const float* xg = x + tile * TOK_PER_WG * DN;
  float* og = out + tile * TOK_PER_WG * DN;

  // ---- stage x tile into LDS (async DMA path when available) ----
  // 32 rows * 256 chunks(16B) = 8192 chunks; 64 threads -> 128 iterations
#if USE_ASYNC_LDS
#pragma unroll 4
  for (int i = 0; i < 128; ++i) {
    int c = i * 64 + t;
    int row = c >> 8;
    int col4 = c & 255;
    const float* g = xg + row * DN + col4 * 4;
    float* l = s_x + row * XPITCH + col4 * 4;
    __builtin_amdgcn_global_load_async_to_lds_b128(
        (__attribute__((address_space(1))) v4i_vs*)(g),
        (__attribute__((address_space(3))) v4i_vs*)(l), 0, 0);
  }
#if __has_builtin(__builtin_amdgcn_s_wait_asynccnt)
  __builtin_amdgcn_s_wait_asynccnt(0);
#else
  asm volatile("s_wait_asynccnt 0x0" ::: "memory");
#endif
#else
#pragma unroll 4
  for (int i = 0; i < 128; ++i) {
    int c = i * 64 + t;
    int row = c >> 8;
    int col4 = c & 255;
    float4 v = *(const float4*)(xg + row * DN + col4 * 4);
    *(float4*)(s_x + row * XPITCH + col4 * 4) = v;
  }
#endif
  __syncthreads();

  // ---- LayerNorm statistics (2 threads per row) ----
  {
    int row = t & 31;
    int half = t >> 5;
    const float* p = s_x + row * XPITCH + half * 512;
    float s = 0.f, q = 0.f;
#pragma unroll 4
    for (int i = 0; i < 512; i += 4) {
      float4 v = *(const float4*)(p + i);
      s += v.x + v.y + v.z + v.w;
      q += v.x * v.x + v.y * v.y + v.z * v.z + v.w * v.w;
    }
    s_ps[t] = s;
    s_pq[t] = q;
  }
  __syncthreads();
  if (t < TOK_PER_WG) {
    float s = s_ps[t] + s_ps[t + 32];
    float q = s_pq[t] + s_pq[t + 32];
    float mu = s * (1.0f / 1024.0f);
    float var = q * (1.0f / 1024.0f) - mu * mu;
    s_mu[t] = mu;
    s_rs[t] = rsqrtf(var + 1e-5f);
  }
  __syncthreads();

  const int row0 = wave * 16;    // 16 tokens per wave
  const int mrow = lane & 15;
  const int khalf = lane >> 4;

  // ---- down GEMM: [16 x 1024] @ [1024 x 48], K-step 4, f32 WMMA ----
  v8f c0 = {0.f, 0.f, 0.f, 0.f, 0.f, 0.f, 0.f, 0.f};
  v8f c1 = c0, c2 = c0;
  {
    const float* arow = s_x + (row0 + mrow) * XPITCH + 2 * khalf;
    const v2f* bp = wgB + lane;
#pragma unroll 2
    for (int ks = 0; ks < 256; ++ks) {
      v2f a = *(const v2f*)(arow + 4 * ks);
      v2f b0v = bp[(0 * 256 + ks) * 32];
      v2f b1v = bp[(1 * 256 + ks) * 32];
      v2f b2v = bp[(2 * 256 + ks) * 32];
      c0 = __builtin_amdgcn_wmma_f32_16x16x4_f32(false, a, false, b0v,
                                                 (short)0, c0, false, false);
      c1 = __builtin_amdgcn_wmma_f32_16x16x4_f32(false, a, false, b1v,
                                                 (short)0, c1, false, false);
      c2 = __builtin_amdgcn_wmma_f32_16x16x4_f32(false, a, false, b2v,
                                                 (short)0, c2, false, false);
    }
  }
  // fold LayerNorm affine + GELU, write h1 to LDS
  {
    float s1v0 = s1g[0 * 16 + mrow], s2v0 = s2g[0 * 16 + mrow];
    float s1v1 = s1g[1 * 16 + mrow], s2v1 = s2g[1 * 16 + mrow];
    float s1v2 = s1g[2 * 16 + mrow], s2v2 = s2g[2 * 16 + mrow];
#pragma unroll
    for (int i = 0; i < 8; ++i) {
      int m = row0 + i + 8 * khalf;
      float mu = s_mu[m], rs = s_rs[m];
      float z0 = rs * (c0[i] - mu * s2v0) + s1v0;
      float z1 = rs * (c1[i] - mu * s2v1) + s1v1;
      float z2 = rs * (c2[i] - mu * s2v2) + s1v2;
      s_h1[m * HPITCH + 0 * 16 + mrow] = gelu_exact(z0);
      s_h1[m * HPITCH + 1 * 16 + mrow] = gelu_exact(z1);
      s_h1[m * HPITCH + 2 * 16 + mrow] = gelu_exact(z2);
    }
  }

  // ---- middle GEMM: [16 x 48] @ fused M [48 x 48] + b0, then GELU ----
  {
    float b00 = b0g[0 * 16 + mrow];
    float b01 = b0g[1 * 16 + mrow];
    float b02 = b0g[2 * 16 + mrow];
    v8f d0, d1, d2;
#pragma unroll
    for (int i = 0; i < 8; ++i) { d0[i] = b00; d1[i] = b01; d2[i] = b02; }
    const float* arow = s_h1 + (row0 + mrow) * HPITCH + 2 * khalf;
    const v2f* mb = midB + lane;
#pragma unroll
    for (int ks = 0; ks < 12; ++ks) {
      v2f a = *(const v2f*)(arow + 4 * ks);
      d0 = __builtin_amdgcn_wmma_f32_16x16x4_f32(false, a, false,
                                                 mb[(0 * 12 + ks) * 32],
                                                 (short)0, d0, false, false);
      d1 = __builtin_amdgcn_wmma_f32_16x16x4_f32(false, a, false,
                                                 mb[(1 * 12 + ks) * 32],
                                                 (short)0, d1, false, false);
      d2 = __builtin_amdgcn_wmma_f32_16x16x4_f32(false, a, false,
                                                 mb[(2 * 12 + ks) * 32],
                                                 (short)0, d2, false, false);
    }
#pragma unroll
    for (int i = 0; i < 8; ++i) {
      int m = row0 + i + 8 * khalf;
      s_h2[m * HPITCH + 0 * 16 + mrow] = gelu_exact(d0[i]);
      s_h2[m * HPITCH + 1 * 16 + mrow] = gelu_exact(d1[i]);
      s_h2[m * HPITCH + 2 * 16 + mrow] = gelu_exact(d2[i]);
    }
  }

  // ---- up GEMM: [16 x 48] @ [48 x 1024] + residual epilogue ----
  {
    const float scale = *scale_p;
    // hoist A fragments (whole K fits in 12 float2 = 24 VGPRs)
    v2f a2[12];
    const float* arow = s_h2 + (row0 + mrow) * HPITCH + 2 * khalf;
#pragma unroll
    for (int ks = 0; ks < 12; ++ks) a2[ks] = *(const v2f*)(arow + 4 * ks);

    for (int nt = 0; nt < 64; ++nt) {
      v8f c = {0.f, 0.f, 0.f, 0.f, 0.f, 0.f, 0.f, 0.f};
      const v2f* ub = upB + nt * 12 * 32 + lane;
#pragma unroll
      for (int ks = 0; ks < 12; ++ks) {
        c = __builtin_amdgcn_wmma_f32_16x16x4_f32(false, a2[ks], false,
                                                  ub[ks * 32], (short)0, c,
                                                  false, false);
      }
      int d = nt * 16 + mrow;
#pragma unroll
      for (int i = 0; i < 8; ++i) {
        int m = row0 + i + 8 * khalf;
        og[m * DN + d] = s_x[m * XPITCH + d] + scale * c[i];
      }
    }
  }
}

// ---------------- launch ----------------

extern "C" void kernel_launch(void* const* d_in, const int* in_sizes, int n_in,
                              void* d_out, int out_size, void* d_ws,
                              size_t ws_size, hipStream_t stream) {
  const float* x = (const float*)d_in[0];
  const float* gamma = (const float*)d_in[1];
  const float* beta = (const float*)d_in[2];
  const float* w_down = (const float*)d_in[3];
  const float* w_to_mv = (const float*)d_in[4];
  const float* w_equi = (const float*)d_in[5];
  const float* b_equi = (const float*)d_in[6];
  const float* w_from_mv = (const float*)d_in[7];
  const float* w_up = (const float*)d_in[8];
  const float* scale = (const float*)d_in[9];
  float* out = (float*)d_out;

  char* ws = (char*)d_ws;
  v2f* wgB = (v2f*)(ws + 0);            // 196608 B
  v2f* upB = (v2f*)(ws + 196608);       // 196608 B
  v2f* midB = (v2f*)(ws + 393216);      // 9216 B
  float* s1 = (float*)(ws + 402432);    // 192 B
  float* s2 = (float*)(ws + 402624);    // 192 B
  float* b0 = (float*)(ws + 402816);    // 192 B

  pack_down<<<96, 256, 0, stream>>>(gamma, w_down, wgB);
  pack_up<<<96, 256, 0, stream>>>(w_up, upB);
  pack_mid<<<5, 256, 0, stream>>>(w_to_mv, w_from_mv, w_equi, b_equi, midB, b0);
  pack_sums<<<1, 64, 0, stream>>>(gamma, beta, w_down, s1, s2);

  const size_t smem =
      (size_t)(TOK_PER_WG * XPITCH + 2 * TOK_PER_WG * HPITCH + 2 * TOK_PER_WG +
               2 * 64) * sizeof(float);  // 145,664 bytes -> 2 WGs per WGP
  (void)hipFuncSetAttribute((const void*)adapter_main,
                            hipFuncAttributeMaxDynamicSharedMemorySize,
                            (int)smem);
  adapter_main<<<1024, 64, smem, stream>>>(x, s1, s2, wgB, midB, b0, upB, scale,
                                           out);
}